// MCT_40836549050921
// MI455X (gfx1250) — compile-verified
//
#include <hip/hip_runtime.h>
#include <math.h>
#include <stdint.h>

#define N 1024
#define DIM 128
#define NP 1024          // padded solve dimension (row/col 1023 zero)
#define NM (N - 1)       // 1023
#define LAMB 0.9f
#define GSIG 15.0f

typedef __attribute__((ext_vector_type(2))) float v2f;
typedef __attribute__((ext_vector_type(8))) float v8f;
typedef __attribute__((ext_vector_type(4))) unsigned int v4u;
typedef __attribute__((ext_vector_type(8))) int v8i;
typedef __attribute__((ext_vector_type(4))) int v4i;

// ---------------------------------------------------------------------------
// TDM: issue a 2D tile load (row-major fp32 tensor -> LDS), D# per ISA 8.3/8.4.
//   width/height in elements; row_stride = tensor row stride (elements).
//   Optional LDS padding: pad_int_code (3b: 1<<c 8B-units between pads),
//   pad_amt_code (7b: code+1 dwords inserted).
// This toolchain exposes the 6-arg builtin: (v4u, v8i, v4i, v4i, v8i, i32).
// ---------------------------------------------------------------------------
__device__ __forceinline__ void tdm_load_2d(unsigned lds_off, const float* gptr,
                                            unsigned width, unsigned height,
                                            unsigned tensor_w, unsigned tensor_h,
                                            unsigned row_stride,
                                            int pad_en, unsigned pad_int_code,
                                            unsigned pad_amt_code) {
    unsigned long long ga = (unsigned long long)(uintptr_t)gptr;
    v4u g0;
    g0.x = 1u;                                         // count=1, user mode
    g0.y = lds_off;                                    // lds_addr (bytes)
    g0.z = (unsigned)(ga & 0xFFFFFFFFu);               // global_addr[31:0]
    g0.w = (unsigned)((ga >> 32) & 0x01FFFFFFu)        // global_addr[56:32]
         | (2u << 30);                                 // type=2 ("image")
    unsigned d0 = (2u << 16);                          // data_size = 4 bytes
    if (pad_en)
        d0 |= (1u << 20) | (pad_int_code << 22) | (pad_amt_code << 25);
    v8i g1;
    g1[0] = (int)d0;                                   // wg_mask=0 (not in cluster)
    g1[1] = (int)((tensor_w & 0xFFFFu) << 16);         // tensor_dim0[15:0]
    g1[2] = (int)(((tensor_w >> 16) & 0xFFFFu) | ((tensor_h & 0xFFFFu) << 16));
    g1[3] = (int)(((tensor_h >> 16) & 0xFFFFu) | ((width & 0xFFFFu) << 16)); // tile_dim0
    g1[4] = (int)(height & 0xFFFFu);                   // tile_dim1; tile_dim2=0
    g1[5] = (int)row_stride;                           // tensor_dim0_stride[31:0]
    g1[6] = 0;
    g1[7] = 0;
    v4i z4 = {0, 0, 0, 0};
    v8i z8 = {0, 0, 0, 0, 0, 0, 0, 0};
    __builtin_amdgcn_tensor_load_to_lds(g0, g1, z4, z4, z8, 0);
}

// ---------------------------------------------------------------------------
// Dense layer: C[1024x128] = A[1024x128] @ W[128x128]^T + bias
// One wave per 16x16 tile, fp32 WMMA, K-step 4. (Tiny: 0.5 MB traffic.)
// ---------------------------------------------------------------------------
__global__ void k_dense(const float* __restrict__ A, const float* __restrict__ W,
                        const float* __restrict__ bias, float* __restrict__ C) {
    const int tm   = (blockIdx.x >> 3) * 16;
    const int tn   = (blockIdx.x & 7) * 16;
    const int lane = threadIdx.x;
    const int half = lane >> 4;
    const int l    = lane & 15;

    v8f acc = {};
    for (int k0 = 0; k0 < DIM; k0 += 4) {
        const int ka = k0 + half * 2;
        v2f a = *(const v2f*)(A + (tm + l) * DIM + ka);      // A[m][k]
        v2f b = *(const v2f*)(W + (tn + l) * DIM + ka);      // B[k][n] = W[n][k]
        acc = __builtin_amdgcn_wmma_f32_16x16x4_f32(false, a, false, b,
                                                    (short)0, acc, false, false);
    }
    const int   n  = tn + l;
    const float bn = bias[n];
    const int   mb = tm + half * 8;
#pragma unroll
    for (int r = 0; r < 8; ++r)
        C[(mb + r) * DIM + n] = acc[r] + bn;
}

// out[i] = sum_k src[i][k]^2
__global__ void k_rowsumsq(const float* __restrict__ src, float* __restrict__ out) {
    __shared__ float red[DIM];
    const int i = blockIdx.x, t = threadIdx.x;
    const float v = src[i * DIM + t];
    red[t] = v * v;
    __syncthreads();
    for (int w = DIM / 2; w > 0; w >>= 1) {
        if (t < w) red[t] += red[t + w];
        __syncthreads();
    }
    if (t == 0) out[i] = red[0];
}

// ---------------------------------------------------------------------------
// Gram + kernel epilogue, 32x32 register tile per wave (2x2 WMMA accs):
//   scores = x@x^T ; S = scores/mag[n] ; dist=(S-1)^2*magf[n] (0 diag)
//   Aout = exp(-0.5*dist/sigma^2)
// ---------------------------------------------------------------------------
__global__ void k_gram(const float* __restrict__ x, const float* __restrict__ mag,
                       const float* __restrict__ magf, float* __restrict__ Aout) {
    const int tm   = (blockIdx.x >> 5) * 32;   // 32x32 tiles over 1024
    const int tn   = (blockIdx.x & 31) * 32;
    const int lane = threadIdx.x;
    const int half = lane >> 4;
    const int l    = lane & 15;

    v8f a00 = {}, a01 = {}, a10 = {}, a11 = {};
    for (int k0 = 0; k0 < DIM; k0 += 4) {
        const int ka = k0 + half * 2;
        v2f fa0 = *(const v2f*)(x + (tm + l) * DIM + ka);
        v2f fa1 = *(const v2f*)(x + (tm + 16 + l) * DIM + ka);
        v2f fb0 = *(const v2f*)(x + (tn + l) * DIM + ka);
        v2f fb1 = *(const v2f*)(x + (tn + 16 + l) * DIM + ka);
        a00 = __builtin_amdgcn_wmma_f32_16x16x4_f32(false, fa0, false, fb0, (short)0, a00, false, false);
        a01 = __builtin_amdgcn_wmma_f32_16x16x4_f32(false, fa0, false, fb1, (short)0, a01, false, false);
        a10 = __builtin_amdgcn_wmma_f32_16x16x4_f32(false, fa1, false, fb0, (short)0, a10, false, false);
        a11 = __builtin_amdgcn_wmma_f32_16x16x4_f32(false, fa1, false, fb1, (short)0, a11, false, false);
    }
    const int   n0 = tn + l,        n1 = tn + 16 + l;
    const float r0 = 1.0f / mag[n0], r1 = 1.0f / mag[n1];
    const float f0 = magf[n0],       f1 = magf[n1];
    const float sc = -0.5f / (GSIG * GSIG);
#pragma unroll
    for (int r = 0; r < 8; ++r) {
        const int m0 = tm + half * 8 + r;
        const int m1 = m0 + 16;
        float S, d;
        S = a00[r] * r0; d = (m0 == n0) ? 0.f : (S - 1.f) * (S - 1.f) * f0;
        Aout[m0 * N + n0] = expf(sc * d);
        S = a01[r] * r1; d = (m0 == n1) ? 0.f : (S - 1.f) * (S - 1.f) * f1;
        Aout[m0 * N + n1] = expf(sc * d);
        S = a10[r] * r0; d = (m1 == n0) ? 0.f : (S - 1.f) * (S - 1.f) * f0;
        Aout[m1 * N + n0] = expf(sc * d);
        S = a11[r] * r1; d = (m1 == n1) ? 0.f : (S - 1.f) * (S - 1.f) * f1;
        Aout[m1 * N + n1] = expf(sc * d);
    }
}

// total[i] = clip(sum_j A[i,j] - A[i,i], 1e-10)
__global__ void k_total(const float* __restrict__ Am, float* __restrict__ total) {
    __shared__ float red[256];
    const int i = blockIdx.x, t = threadIdx.x;
    float s = 0.0f;
    for (int j = t; j < N; j += 256) s += Am[i * N + j];
    red[t] = s;
    __syncthreads();
    for (int w = 128; w > 0; w >>= 1) {
        if (t < w) red[t] += red[t + w];
        __syncthreads();
    }
    if (t == 0) total[i] = fmaxf(red[0] - Am[i * N + i], 1e-10f);
}

// B (padded 1024x1024) = lambda * T[1:,1:], zero diag / pad
__global__ void k_buildB(const float* __restrict__ Am, const float* __restrict__ total,
                         float* __restrict__ B) {
    const int idx = blockIdx.x * 256 + threadIdx.x;
    const int i = idx >> 10, j = idx & (NP - 1);
    float v = 0.0f;
    if (i < NM && j < NM && i != j)
        v = LAMB * Am[(i + 1) * N + (j + 1)] / total[i + 1];
    B[idx] = v;
}

// p0 (padded) = T[0, 1:]
__global__ void k_buildP0(const float* __restrict__ Am, const float* __restrict__ total,
                          float* __restrict__ p0) {
    const int j = blockIdx.x * 256 + threadIdx.x;
    p0[j] = (j < NM) ? Am[j + 1] / total[0] : 0.0f;
}

// out = p + B @ p   (one row per block)
__global__ void k_matvec(const float* __restrict__ B, const float* __restrict__ p,
                         float* __restrict__ out) {
    __shared__ float red[256];
    const int i = blockIdx.x, t = threadIdx.x;
    float s = 0.0f;
    for (int j = t; j < NP; j += 256) s += B[i * NP + j] * p[j];
    red[t] = s;
    __syncthreads();
    for (int w = 128; w > 0; w >>= 1) {
        if (t < w) red[t] += red[t + w];
        __syncthreads();
    }
    if (t == 0) out[i] = p[i] + red[0];
}

// ---------------------------------------------------------------------------
// Cm = Bm @ Bm (1024^3, fp32 WMMA). 64x64 macro-tile per 128-thread workgroup
// (4 waves, each a 32x32 register tile). K-chunks of 32 staged into LDS by the
// Tensor Data Mover (double-buffered): wave 0 issues tensor_load_to_lds for
// A(64x32, LDS row stride 34 via TDM pad -> conflict-free) and B(32x64),
// waits s_wait_tensorcnt 0, then the workgroup barrier releases all 4 waves.
// ---------------------------------------------------------------------------
#define ASTRIDE 34                       // 32 + 2 pad dwords per row
__global__ void k_square(const float* __restrict__ Bm, float* __restrict__ Cm) {
    __shared__ float As[2][64 * ASTRIDE];
    __shared__ float Bs[2][32 * 64];

    const int tmWG = (blockIdx.x >> 4) * 64;   // 16x16 macro-tiles
    const int tnWG = (blockIdx.x & 15) * 64;
    const int wid  = threadIdx.x >> 5;         // wave 0..3
    const int wr   = wid >> 1;                 // wave row (0/1)
    const int wc   = wid & 1;                  // wave col (0/1)
    const int lane = threadIdx.x & 31;
    const int half = lane >> 4;
    const int l    = lane & 15;

    // prologue: chunk 0 -> buffer 0  (wave 0 drives the TDM)
    if (threadIdx.x < 32) {
        tdm_load_2d((unsigned)(uintptr_t)&As[0][0], Bm + tmWG * NP,
                    32, 64, NP, NP, NP, /*pad*/1, /*1<<4 x8B=32dw*/4, /*2dw*/1);
        tdm_load_2d((unsigned)(uintptr_t)&Bs[0][0], Bm + tnWG,
                    64, 32, NP, NP, NP, 0, 0, 0);
    }

    v8f a00 = {}, a01 = {}, a10 = {}, a11 = {};
    int buf = 0;
    for (int c = 0; c < 32; ++c) {
        if (threadIdx.x < 32) __builtin_amdgcn_s_wait_tensorcnt(0);
        __syncthreads();                       // chunk c visible to all waves
        if (c + 1 < 32 && threadIdx.x < 32) {  // prefetch chunk c+1 -> other buffer
            const int k1 = (c + 1) * 32;
            tdm_load_2d((unsigned)(uintptr_t)&As[buf ^ 1][0], Bm + tmWG * NP + k1,
                        32, 64, NP, NP, NP, 1, 4, 1);
            tdm_load_2d((unsigned)(uintptr_t)&Bs[buf ^ 1][0], Bm + k1 * NP + tnWG,
                        64, 32, NP, NP, NP, 0, 0, 0);
        }
        const float* Ab = &As[buf][wr * 32 * ASTRIDE];
        const float* Bb = &Bs[buf][wc * 32];
#pragma unroll
        for (int kk = 0; kk < 32; kk += 4) {
            const int ka = kk + half * 2;
            v2f fa0 = *(const v2f*)(Ab + l * ASTRIDE + ka);
            v2f fa1 = *(const v2f*)(Ab + (l + 16) * ASTRIDE + ka);
            v2f fb0, fb1;
            fb0.x = Bb[(ka + 0) * 64 + l];      fb0.y = Bb[(ka + 1) * 64 + l];
            fb1.x = Bb[(ka + 0) * 64 + 16 + l]; fb1.y = Bb[(ka + 1) * 64 + 16 + l];
            a00 = __builtin_amdgcn_wmma_f32_16x16x4_f32(false, fa0, false, fb0, (short)0, a00, false, false);
            a01 = __builtin_amdgcn_wmma_f32_16x16x4_f32(false, fa0, false, fb1, (short)0, a01, false, false);
            a10 = __builtin_amdgcn_wmma_f32_16x16x4_f32(false, fa1, false, fb0, (short)0, a10, false, false);
            a11 = __builtin_amdgcn_wmma_f32_16x16x4_f32(false, fa1, false, fb1, (short)0, a11, false, false);
        }
        __syncthreads();                       // reads of buf done before reuse
        buf ^= 1;
    }

    const int mb = tmWG + wr * 32;
    const int nb = tnWG + wc * 32;
#pragma unroll
    for (int r = 0; r < 8; ++r) {
        const int m0 = mb + half * 8 + r;
        const int m1 = m0 + 16;
        Cm[m0 * NP + nb + l]      = a00[r];
        Cm[m0 * NP + nb + 16 + l] = a01[r];
        Cm[m1 * NP + nb + l]      = a10[r];
        Cm[m1 * NP + nb + 16 + l] = a11[r];
    }
}

// P = (1-lambda)*p; P -= mean; relu; *100; softmax -> out (1023 elems)
__global__ void k_finalize(const float* __restrict__ p, float* __restrict__ out) {
    __shared__ float red[1024];
    const int t = threadIdx.x;
    const float v = (t < NM) ? (1.0f - LAMB) * p[t] : 0.0f;
    red[t] = v;
    __syncthreads();
    for (int w = 512; w > 0; w >>= 1) {
        if (t < w) red[t] += red[t + w];
        __syncthreads();
    }
    const float mean = red[0] / (float)NM;
    __syncthreads();
    float q = v - mean;
    q = (q < 0.0f) ? 0.0f : q;
    q *= 100.0f;
    red[t] = (t < NM) ? q : -1e30f;
    __syncthreads();
    for (int w = 512; w > 0; w >>= 1) {
        if (t < w) red[t] = fmaxf(red[t], red[t + w]);
        __syncthreads();
    }
    const float mx = red[0];
    __syncthreads();
    const float e = (t < NM) ? expf(q - mx) : 0.0f;
    red[t] = e;
    __syncthreads();
    for (int w = 512; w > 0; w >>= 1) {
        if (t < w) red[t] += red[t + w];
        __syncthreads();
    }
    if (t < NM) out[t] = e / red[0];
}

// ---------------------------------------------------------------------------
extern "C" void kernel_launch(void* const* d_in, const int* in_sizes, int n_in,
                              void* d_out, int out_size, void* d_ws, size_t ws_size,
                              hipStream_t stream) {
    const float* fin = (const float*)d_in[0];
    const float* w1 = (const float*)d_in[1]; const float* b1 = (const float*)d_in[2];
    const float* w2 = (const float*)d_in[3]; const float* b2 = (const float*)d_in[4];
    const float* w3 = (const float*)d_in[5]; const float* b3 = (const float*)d_in[6];
    const float* w4 = (const float*)d_in[7]; const float* b4 = (const float*)d_in[8];
    float* outp = (float*)d_out;

    float* ws    = (float*)d_ws;
    float* t0    = ws;               // N*DIM
    float* t1    = t0 + N * DIM;     // N*DIM
    float* mag   = t1 + N * DIM;     // N
    float* magf  = mag + N;          // N
    float* total = magf + N;         // N
    float* pa    = total + N;        // NP
    float* pb    = pa + NP;          // NP
    float* Amat  = pb + NP;          // N*N
    float* B0    = Amat + N * N;     // NP*NP
    float* B1    = B0 + NP * NP;     // NP*NP   (~13.7 MB total, L2-resident)

    // 4 dense layers (order: w2, w3, w4 -> f ; then w1 -> x)
    k_dense<<<512, 32, 0, stream>>>(fin, w2, b2, t0);
    k_dense<<<512, 32, 0, stream>>>(t0, w3, b3, t1);
    k_dense<<<512, 32, 0, stream>>>(t1, w4, b4, t0);   // t0 = f
    k_dense<<<512, 32, 0, stream>>>(t0, w1, b1, t1);   // t1 = x

    k_rowsumsq<<<N, DIM, 0, stream>>>(t1, mag);        // ||x_j||^2
    k_rowsumsq<<<N, DIM, 0, stream>>>(t0, magf);       // ||f_j||^2

    k_gram<<<1024, 32, 0, stream>>>(t1, mag, magf, Amat);
    k_total<<<N, 256, 0, stream>>>(Amat, total);
    k_buildB<<<NP * NP / 256, 256, 0, stream>>>(Amat, total, B0);
    k_buildP0<<<NP / 256, 256, 0, stream>>>(Amat, total, pa);

    // (I-B)^-1 p0 = prod_k (I + B^{2^k}) p0 : 8 matvecs, 7 squarings
    float* Bcur = B0; float* Bnxt = B1;
    float* pcur = pa; float* pnxt = pb;
    for (int k = 0; k < 8; ++k) {
        k_matvec<<<NP, 256, 0, stream>>>(Bcur, pcur, pnxt);
        { float* tmp = pcur; pcur = pnxt; pnxt = tmp; }
        if (k < 7) {
            k_square<<<256, 128, 0, stream>>>(Bcur, Bnxt);
            float* tmp = Bcur; Bcur = Bnxt; Bnxt = tmp;
        }
    }

    k_finalize<<<1, 1024, 0, stream>>>(pcur, outp);
}